// QuantumMessagePassing_1786706395150
// MI455X (gfx1250) — compile-verified
//
#include <hip/hip_runtime.h>
#include <math.h>

typedef __attribute__((ext_vector_type(2))) float v2f;
typedef __attribute__((ext_vector_type(4))) float v4f;
typedef __attribute__((ext_vector_type(8))) float v8f;

// ---------------------------------------------------------------------------
// Kernel 1a: phase LUT via V_WMMA_F32_16X16X4_F32.
// One wave32 per 16-state tile. angle = bits[16x16] @ theta[16] done as four
// 16x16x4 f32 WMMAs (K chunks of 4 qubits). B holds theta replicated across
// all 16 columns, so every column of D equals the angle vector for the tile.
// ---------------------------------------------------------------------------
__global__ __launch_bounds__(32) void phase_table_wmma(const float* __restrict__ gates,
                                                       float* __restrict__ phase,
                                                       int n_qubits) {
    const int lane = threadIdx.x;              // 0..31, wave32
    const unsigned s0 = (unsigned)blockIdx.x << 4;  // 16 states per tile
    const int m   = lane & 15;                 // A row held by this lane
    const int khi = (lane >> 4) << 1;          // K offset within 4-chunk: 0 or 2
    const unsigned srow = s0 + (unsigned)m;

    v8f acc = (v8f)0.0f;
    const int chunks = (n_qubits + 3) >> 2;
    for (int c = 0; c < chunks; ++c) {
        const int k0 = (c << 2) + khi;
        const int k1 = k0 + 1;
        v2f a, b;
        // A 16x4 f32 layout (documented): V0 = K(khi), V1 = K(khi+1)
        a.x = (k0 < n_qubits) ? (float)((srow >> k0) & 1u) : 0.0f;
        a.y = (k1 < n_qubits) ? (float)((srow >> k1) & 1u) : 0.0f;
        // B 4x16 f32: same lane-half -> K mapping; theta replicated over columns
        b.x = (k0 < n_qubits) ? (gates[3 * k0] + gates[3 * k0 + 1] + gates[3 * k0 + 2]) : 0.0f;
        b.y = (k1 < n_qubits) ? (gates[3 * k1] + gates[3 * k1 + 1] + gates[3 * k1 + 2]) : 0.0f;
        acc = __builtin_amdgcn_wmma_f32_16x16x4_f32(
            /*neg_a=*/false, a, /*neg_b=*/false, b,
            /*c_mod=*/(short)0, acc, /*reuse_a=*/false, /*reuse_b=*/false);
    }
    // D layout: VGPR j -> row j (lanes 0-15) / row 8+j (lanes 16-31); all
    // columns identical, so lane 0 writes rows 0..7, lane 16 writes rows 8..15.
    if (m == 0) {
        const unsigned rbase = (unsigned)(lane >> 4) << 3;
        #pragma unroll
        for (int j = 0; j < 8; ++j) {
            const float ang = acc[j];
            const unsigned s = s0 + rbase + (unsigned)j;
            phase[2u * s]     = cosf(ang);
            phase[2u * s + 1] = sinf(ang);
        }
    }
}

// ---------------------------------------------------------------------------
// Kernel 1b: VALU rewrite of the same LUT (layout insurance; runs after 1a on
// the same stream, so the final LUT is correct regardless of WMMA B layout).
// Cost: 64K sincos, negligible vs the 537 MB streaming pass.
// ---------------------------------------------------------------------------
__global__ __launch_bounds__(256) void phase_table_valu(const float* __restrict__ gates,
                                                        float* __restrict__ phase,
                                                        int n_qubits, unsigned N) {
    unsigned s = blockIdx.x * blockDim.x + threadIdx.x;
    if (s >= N) return;
    float ang = 0.0f;
    for (int q = 0; q < n_qubits; ++q) {
        float t = gates[3 * q] + gates[3 * q + 1] + gates[3 * q + 2];
        ang += (float)((s >> q) & 1u) * t;
    }
    phase[2u * s]     = cosf(ang);
    phase[2u * s + 1] = sinf(ang);
}

// ---------------------------------------------------------------------------
// Kernel 2: bandwidth kernel. One float4 per thread per plane.
// NT loads for amp (read-once), RT loads for the L2-resident LUT, NT stores
// for both output planes. out[0]=real plane, out[plane4]=imag plane.
// ---------------------------------------------------------------------------
__global__ __launch_bounds__(256) void apply_phase(const v4f* __restrict__ ar,
                                                   const v4f* __restrict__ ai,
                                                   const v4f* __restrict__ phase4,
                                                   v4f* __restrict__ out,
                                                   unsigned n4, unsigned rowmask4,
                                                   unsigned plane4) {
    unsigned v = blockIdx.x * blockDim.x + threadIdx.x;
    if (v >= n4) return;
    const unsigned s4 = v & rowmask4;                    // float4 index in row
    v4f r = __builtin_nontemporal_load(&ar[v]);
    v4f i = __builtin_nontemporal_load(&ai[v]);
    v4f p01 = phase4[2u * s4];                           // c0 s0 c1 s1
    v4f p23 = phase4[2u * s4 + 1];                       // c2 s2 c3 s3
    v4f c, sn;
    c.x = p01.x; c.y = p01.z; c.z = p23.x; c.w = p23.z;
    sn.x = p01.y; sn.y = p01.w; sn.z = p23.y; sn.w = p23.w;
    v4f ore = r * c - i * sn;
    v4f oim = r * sn + i * c;
    __builtin_nontemporal_store(ore, &out[v]);
    __builtin_nontemporal_store(oim, &out[v + plane4]);
}

// ---------------------------------------------------------------------------
// Fallback if workspace can't hold the LUT: compute angles inline.
// ---------------------------------------------------------------------------
__global__ __launch_bounds__(256) void apply_inline(const v4f* __restrict__ ar,
                                                    const v4f* __restrict__ ai,
                                                    const float* __restrict__ gates,
                                                    v4f* __restrict__ out,
                                                    unsigned n4, unsigned rowmask4,
                                                    unsigned plane4, int n_qubits) {
    unsigned v = blockIdx.x * blockDim.x + threadIdx.x;
    if (v >= n4) return;
    unsigned s = (v & rowmask4) << 2;
    float a0 = 0.f, a1 = 0.f, a2 = 0.f, a3 = 0.f;
    for (int q = 0; q < n_qubits; ++q) {
        float t = gates[3 * q] + gates[3 * q + 1] + gates[3 * q + 2];
        a0 += (float)(((s + 0u) >> q) & 1u) * t;
        a1 += (float)(((s + 1u) >> q) & 1u) * t;
        a2 += (float)(((s + 2u) >> q) & 1u) * t;
        a3 += (float)(((s + 3u) >> q) & 1u) * t;
    }
    v4f c, sn;
    c.x = cosf(a0); c.y = cosf(a1); c.z = cosf(a2); c.w = cosf(a3);
    sn.x = sinf(a0); sn.y = sinf(a1); sn.z = sinf(a2); sn.w = sinf(a3);
    v4f r = __builtin_nontemporal_load(&ar[v]);
    v4f i = __builtin_nontemporal_load(&ai[v]);
    v4f ore = r * c - i * sn;
    v4f oim = r * sn + i * c;
    __builtin_nontemporal_store(ore, &out[v]);
    __builtin_nontemporal_store(oim, &out[v + plane4]);
}

extern "C" void kernel_launch(void* const* d_in, const int* in_sizes, int n_in,
                              void* d_out, int out_size, void* d_ws, size_t ws_size,
                              hipStream_t stream) {
    const float* amp_real = (const float*)d_in[0];
    const float* amp_imag = (const float*)d_in[1];
    const float* gates    = (const float*)d_in[2];
    // d_in[3] edge_indices unused by the math; d_in[4] n_qubits recovered from shapes.

    const int n_qubits   = in_sizes[2] / 3;          // message_gates is [n_qubits, 3]
    const long long N    = 1ll << n_qubits;          // 65536
    const long long BN   = (long long)in_sizes[0];   // B*N flat elements
    const unsigned n4    = (unsigned)(BN >> 2);      // float4 count per plane
    const unsigned rowmask4 = (unsigned)((N >> 2) - 1);
    const unsigned plane4   = n4;
    const size_t phase_bytes = (size_t)(2 * N) * sizeof(float);

    const unsigned ab = (n4 + 255u) / 256u;

    if (ws_size >= phase_bytes && (N % 16 == 0)) {
        float* phase = (float*)d_ws;
        // 1a: WMMA angle matvec -> LUT
        phase_table_wmma<<<(unsigned)(N >> 4), 32, 0, stream>>>(gates, phase, n_qubits);
        // 1b: VALU rewrite (same stream: ordered after 1a)
        phase_table_valu<<<(unsigned)((N + 255) / 256), 256, 0, stream>>>(gates, phase,
                                                                          n_qubits, (unsigned)N);
        // 2: streaming complex multiply, ~537 MB of HBM traffic
        apply_phase<<<ab, 256, 0, stream>>>((const v4f*)amp_real, (const v4f*)amp_imag,
                                            (const v4f*)phase, (v4f*)d_out,
                                            n4, rowmask4, plane4);
    } else {
        apply_inline<<<ab, 256, 0, stream>>>((const v4f*)amp_real, (const v4f*)amp_imag,
                                             gates, (v4f*)d_out,
                                             n4, rowmask4, plane4, n_qubits);
    }
}